// H_UOT_Observer_4904852652411
// MI455X (gfx1250) — compile-verified
//
#include <hip/hip_runtime.h>
#include <hip/hip_bf16.h>
#include <math.h>

typedef __attribute__((ext_vector_type(2))) float v2f;
typedef __attribute__((ext_vector_type(8))) float v8f;

#define REGC  0.05f
#define POWC  (1.0f / 1.05f)     // ALPHA/(ALPHA+REG)
#define EPSC  1e-6f

#define BB 256
#define NN 1024
#define CC 512
#define NT 256      // threads per block
#define NW 8        // waves per block (wave32)

// all-thread block reduction; `red` is a 256-float LDS buffer.
// Leading barrier protects the previous use of `red` and publishes
// any prior per-wave LDS writes.
__device__ __forceinline__ float block_reduce_sum(float v, float* red) {
    const int tid = threadIdx.x;
    __syncthreads();
    red[tid] = v;
    __syncthreads();
    #pragma unroll
    for (int s = NT / 2; s > 0; s >>= 1) {
        if (tid < s) red[tid] += red[tid + s];
        __syncthreads();
    }
    return red[0];
}

__global__ __launch_bounds__(NT) void huot_observer_kernel(
    const float* __restrict__ prompt,   // (B,1,C)
    const float* __restrict__ vis,      // (B,N,C)
    const float* __restrict__ dust,     // (1,1,C)
    float* __restrict__ out)            // p_obs (B,C) then total_mass (B)
{
    __shared__ float s_p[CC];
    __shared__ float s_dot[NN + 1];
    __shared__ float s_n2[NN + 1];
    __shared__ float s_K[NN + 1];
    __shared__ float red[NT];

    const int tid   = threadIdx.x;
    const int b     = blockIdx.x;
    const int lane  = tid & 31;
    const int wave  = tid >> 5;
    const int m     = lane & 15;     // A-matrix row within tile
    const int khalf = lane >> 4;     // A-matrix K half (K = 2*khalf, 2*khalf+1)

    const float* Vb = vis + (size_t)b * NN * CC;
    const float* Pb = prompt + (size_t)b * CC;

    // ---- prompt into LDS + ||p||^2 ----
    float pp = 0.f;
    for (int c = tid; c < CC; c += NT) {
        float x = Pb[c];
        s_p[c] = x;
        pp += x * x;
    }
    float pnorm = fmaxf(sqrtf(block_reduce_sum(pp, red)), 1e-12f);

    // ---- Phase A: sim dots + row norms via V_WMMA_F32_16X16X4_F32 ----
    // A = 16 rows of V (16x4 f32 fragment), B = prompt chunk replicated
    // across all 16 columns -> every column of D holds the 16 row dots.
    for (int tile = wave; tile < NN / 16; tile += NW) {
        const float* rowp = Vb + (size_t)(tile * 16 + m) * CC + 2 * khalf;
        v8f acc = {};
        float n2 = 0.f;
        #pragma unroll 4
        for (int kb = 0; kb < CC; kb += 4) {
            v2f a;  a.x  = rowp[kb];            a.y  = rowp[kb + 1];
            v2f bm; bm.x = s_p[kb + 2 * khalf]; bm.y = s_p[kb + 2 * khalf + 1];
            acc = __builtin_amdgcn_wmma_f32_16x16x4_f32(
                false, a, false, bm, (short)0, acc, false, false);
            n2 += a.x * a.x + a.y * a.y;
        }
        // lane m holds K-half of row m's norm; pair with lane m+16
        n2 += __shfl_xor(n2, 16, 32);
        if (lane < 16) s_n2[tile * 16 + lane] = n2;
        // D layout: VGPR r -> row r (lanes 0-15) / row r+8 (lanes 16-31)
        if ((lane & 15) == 0) {
            const int rbase = tile * 16 + (khalf ? 8 : 0);
            #pragma unroll
            for (int r = 0; r < 8; ++r) s_dot[rbase + r] = acc[r];
        }
    }

    // ---- dustbin row (n = N) ----
    float dd = 0.f, dn = 0.f;
    for (int c = tid; c < CC; c += NT) {
        float x = dust[c];
        dd += s_p[c] * x;
        dn += x * x;
    }
    float ddot = block_reduce_sum(dd, red);
    float dn2  = block_reduce_sum(dn, red);
    if (tid == 0) { s_dot[NN] = ddot; s_n2[NN] = dn2; }
    __syncthreads();

    // ---- K = exp((sim - 1)/REG) ----
    for (int n = tid; n < NN + 1; n += NT) {
        float denom = pnorm * fmaxf(sqrtf(s_n2[n]), 1e-12f);
        float sim = s_dot[n] / denom;
        s_K[n] = expf((sim - 1.0f) * (1.0f / REGC));
    }
    __syncthreads();

    // ---- Sinkhorn: u is scalar, v elementwise in (K, u) ----
    float u = 0.f;
    #pragma unroll 1
    for (int it = 0; it < 3; ++it) {
        float part = 0.f;
        for (int n = tid; n < NN + 1; n += NT) {
            float K = s_K[n];
            float v = (it == 0) ? 1.0f : powf(u * K + EPSC, -POWC);
            part += K * v;
        }
        float Kv = block_reduce_sum(part, red);
        u = powf(Kv + EPSC, -POWC);   // identical on all threads
    }

    // ---- gamma (excluding dustbin) + total mass ----
    float mp = 0.f;
    for (int n = tid; n < NN; n += NT) {
        float K = s_K[n];
        float g = u * K * powf(u * K + EPSC, -POWC);
        s_K[n] = g;                   // overwrite with gamma
        mp += g;
    }
    float mass = block_reduce_sum(mp, red);   // barrier also publishes gamma
    if (tid == 0) out[(size_t)BB * CC + b] = mass;
    __syncthreads();

    // ---- Phase B: p_obs = gamma^T * V via WMMA ----
    // A = gamma chunk replicated into all 16 rows, B = 4x16 tile of V
    // (coalesced 64B row segments) -> every row of D holds 16 p_obs cols.
    for (int ct = wave; ct < CC / 16; ct += NW) {
        const int cbase = ct * 16;
        const float* colp = Vb + (size_t)(2 * khalf) * CC + cbase + (lane & 15);
        v8f acc = {};
        #pragma unroll 2
        for (int nb = 0; nb < NN; nb += 4) {
            v2f a;  a.x  = s_K[nb + 2 * khalf];      a.y  = s_K[nb + 2 * khalf + 1];
            v2f bm; bm.x = colp[(size_t)nb * CC];    bm.y = colp[(size_t)nb * CC + CC];
            acc = __builtin_amdgcn_wmma_f32_16x16x4_f32(
                false, a, false, bm, (short)0, acc, false, false);
        }
        if (lane < 16) out[(size_t)b * CC + cbase + lane] = acc[0];
    }
}

extern "C" void kernel_launch(void* const* d_in, const int* in_sizes, int n_in,
                              void* d_out, int out_size, void* d_ws, size_t ws_size,
                              hipStream_t stream) {
    (void)in_sizes; (void)n_in; (void)d_ws; (void)ws_size; (void)out_size;
    const float* prompt = (const float*)d_in[0];
    const float* vis    = (const float*)d_in[1];
    const float* dust   = (const float*)d_in[2];
    float* out = (float*)d_out;
    hipLaunchKernelGGL(huot_observer_kernel, dim3(BB), dim3(NT), 0, stream,
                       prompt, vis, dust, out);
}